// LearnedRigidBody_23235773071845
// MI455X (gfx1250) — compile-verified
//
#include <hip/hip_runtime.h>
#include <math.h>

// ---------------------------------------------------------------------------
// Fused LearnedRigidBody forward dynamics for gfx1250 (MI455X, wave32, WMMA).
// One wave (32 threads) processes 16 batch rows end-to-end using
// V_WMMA_F32_16X16X4_F32 for every dense layer (forward, jvp tangents, vjp).
// Hidden activations use the CDNA5 hardware tanh transcendental when exposed.
// ---------------------------------------------------------------------------

typedef float v2f __attribute__((ext_vector_type(2)));
typedef float v8f __attribute__((ext_vector_type(8)));

static constexpr int LSTRIDE   = 132;  // padded LDS row stride (floats): bank-conflict-free
static constexpr int TILE_ROWS = 16;

__device__ __forceinline__ float fast_tanh(float x) {
#if __has_builtin(__builtin_amdgcn_tanhf)
  return __builtin_amdgcn_tanhf(x);            // v_tanh_f32 (CDNA5 TRANS op)
#elif __has_builtin(__builtin_amdgcn_tanh_f32)
  return __builtin_amdgcn_tanh_f32(x);
#else
  return tanhf(x);                             // ocml fallback
#endif
}

// A-matrix slab (16x4 f32): lane L holds row (L&15); VGPR j holds K = 4*ks + j + 2*(L>>4)
__device__ __forceinline__ v2f lds_load_a(const float* ldsA, int ks, int K) {
  const int lane = threadIdx.x & 31;
  const int row  = lane & 15;
  const int col  = 4 * ks + 2 * (lane >> 4);
  v2f a;
  if (col + 1 < K) {
    a = *(const v2f*)(ldsA + row * LSTRIDE + col);   // ds_load_b64 (8B aligned)
  } else {
    a.x = (col < K) ? ldsA[row * LSTRIDE + col] : 0.f;
    a.y = 0.f;
  }
  return a;
}

// B-matrix slab (4x16 f32) from row-major W[K x N]: element (k,n), n = nb + (L&15)
__device__ __forceinline__ v2f g_load_b(const float* __restrict__ W, int ldw,
                                        int ks, int nb, int K, int N) {
  const int lane = threadIdx.x & 31;
  const int n  = nb + (lane & 15);
  const int k0 = 4 * ks + (lane >> 4);
  const int k1 = k0 + 2;
  v2f b;
  b.x = (k0 < K && n < N) ? W[k0 * ldw + n] : 0.f;
  b.y = (k1 < K && n < N) ? W[k1 * ldw + n] : 0.f;
  return b;
}

// Transposed-B slab: logical B(k,n) = W[n*ldw + k]  (for the vjp GEMMs with W^T)
__device__ __forceinline__ v2f g_load_bT(const float* __restrict__ W, int ldw,
                                         int ks, int nb, int K, int N) {
  const int lane = threadIdx.x & 31;
  const int n  = nb + (lane & 15);
  const int k0 = 4 * ks + (lane >> 4);
  const int k1 = k0 + 2;
  v2f b;
  b.x = (k0 < K && n < N) ? W[n * ldw + k0] : 0.f;
  b.y = (k1 < K && n < N) ? W[n * ldw + k1] : 0.f;
  return b;
}

__device__ __forceinline__ v8f wmma4(v2f a, v2f b, v8f c) {
  // D = A(16x4) * B(4x16) + C(16x16), full f32
  return __builtin_amdgcn_wmma_f32_16x16x4_f32(false, a, false, b, (short)0, c, false, false);
}

// EPI: 0 = none, 1 = tanh, 2 = multiply by (1 - h^2) with h read from ldsH (same (m,n) position)
template <int K, int N, int EPI, bool TRANSB>
__device__ __forceinline__ void gemm16(const float* __restrict__ W, int ldw,
                                       const float* __restrict__ bias,
                                       const float* ldsA,
                                       float* ldsOut, int outStride, int outCols,
                                       const float* ldsH) {
  constexpr int NT = (N + 15) / 16;
  constexpr int KS = (K + 3) / 4;
  const int lane = threadIdx.x & 31;

  v8f acc[NT];
#pragma unroll
  for (int nt = 0; nt < NT; ++nt) {
    const int n = nt * 16 + (lane & 15);
    const float b0 = (bias != nullptr && n < N) ? bias[n] : 0.f;  // C[m][n] = bias[n]
#pragma unroll
    for (int r = 0; r < 8; ++r) acc[nt][r] = b0;
  }

  for (int ks = 0; ks < KS; ++ks) {
    const v2f a = lds_load_a(ldsA, ks, K);
#pragma unroll
    for (int nt = 0; nt < NT; ++nt) {
      v2f b;
      if constexpr (TRANSB) b = g_load_bT(W, ldw, ks, nt * 16, K, N);
      else                  b = g_load_b (W, ldw, ks, nt * 16, K, N);
      acc[nt] = wmma4(a, b, acc[nt]);
    }
  }

  // Transposed store of D tiles back to row-major LDS (+ fused epilogue).
  // D layout: element (m,n): vgpr r = m%8, lane = n + 16*(m/8).
#pragma unroll
  for (int nt = 0; nt < NT; ++nt) {
    const int n = nt * 16 + (lane & 15);
    if (n < outCols) {
#pragma unroll
      for (int r = 0; r < 8; ++r) {
        const int m = r + 8 * (lane >> 4);
        float val = acc[nt][r];
        if (EPI == 1) val = fast_tanh(val);
        if (EPI == 2) { const float h = ldsH[m * LSTRIDE + n]; val *= (1.f - h * h); }
        ldsOut[m * outStride + n] = val;
      }
    }
  }
}

struct LrbParams {
  const float *q, *v, *u;
  const float *Wm1, *bm1, *Wm2, *bm2, *Wm3, *bm3;   // mass:  7->128->128->28
  const float *Wp1, *bp1, *Wp2, *bp2, *Wp3, *bp3;   // pot:   7->128->128->1
  const float *Wf1, *bf1, *Wf2, *bf2, *Wf3, *bf3;   // force: 21->128->128->7
  float* out;
  int B;
};

__global__ __launch_bounds__(32) void lrb_fused_kernel(LrbParams p) {
  __shared__ __align__(16) float bufA[TILE_ROWS * LSTRIDE];  // X, then tangent staging
  __shared__ __align__(16) float bufB[TILE_ROWS * LSTRIDE];  // hidden-1 activations
  __shared__ __align__(16) float bufT[TILE_ROWS * LSTRIDE];  // hidden-2 activations / grads
  __shared__ __align__(16) float rawL[TILE_ROWS * 28];       // mass-net raw tril output
  __shared__ __align__(16) float rawD[TILE_ROWS * 28];       // F / G / tangent raw scratch

  const int lane = threadIdx.x & 31;
  const int r0   = blockIdx.x * TILE_ROWS;

  // Prefetch the most-reused weight matrix (Wm2 is streamed 8x per wave) into
  // the cache hierarchy early: emits global_prefetch_b8.
#pragma unroll
  for (int it = 0; it < 16; ++it) {
    __builtin_prefetch(p.Wm2 + (it * 32 + lane) * 32, 0, 3);  // 128B line per lane
  }

  // ---- stage X = [q, v, u] (16 x 21) into bufA ----
  for (int t = lane; t < TILE_ROWS * 21; t += 32) {
    const int m = t / 21, c = t - m * 21;
    int row = r0 + m; if (row >= p.B) row = p.B - 1;
    float val;
    if (c < 7)       val = p.q[row * 7 + c];
    else if (c < 14) val = p.v[row * 7 + (c - 7)];
    else             val = p.u[row * 7 + (c - 14)];
    bufA[m * LSTRIDE + c] = val;
  }

  float vv[7] = {0, 0, 0, 0, 0, 0, 0};
  const bool rowOwner = (lane < 16) && (r0 + lane < p.B);
  if (rowOwner) {
#pragma unroll
    for (int j = 0; j < 7; ++j) vv[j] = p.v[(r0 + lane) * 7 + j];
  }
  float rhs[7] = {0, 0, 0, 0, 0, 0, 0};
  float Cv[7]  = {0, 0, 0, 0, 0, 0, 0};

  // ================= force net: F = MLP([q,v,u]) =================
  gemm16<21, 128, 1, false>(p.Wf1, 128, p.bf1, bufA, bufB, LSTRIDE, 128, nullptr);
  gemm16<128, 128, 1, false>(p.Wf2, 128, p.bf2, bufB, bufT, LSTRIDE, 128, nullptr);
  gemm16<128, 7, 0, false>(p.Wf3, 7, p.bf3, bufT, rawD, 28, 7, nullptr);
  if (rowOwner) {
#pragma unroll
    for (int j = 0; j < 7; ++j) rhs[j] += rawD[lane * 28 + j];
  }

  // ================= potential net: G = dV/dq (reverse mode) =================
  gemm16<7, 128, 1, false>(p.Wp1, 128, p.bp1, bufA, bufB, LSTRIDE, 128, nullptr);
  gemm16<128, 128, 1, false>(p.Wp2, 128, p.bp2, bufB, bufT, LSTRIDE, 128, nullptr);
  // g_z2 = w3 ⊙ (1 - h2^2), in place in bufT
  for (int t = lane; t < TILE_ROWS * 128; t += 32) {
    const int m = t >> 7, c = t & 127;
    const float h = bufT[m * LSTRIDE + c];
    bufT[m * LSTRIDE + c] = p.Wp3[c] * (1.f - h * h);
  }
  // g_z1 = (g_z2 @ W2^T) ⊙ (1 - h1^2), in place; then G = g_z1 @ W1^T
  gemm16<128, 128, 2, true>(p.Wp2, 128, nullptr, bufT, bufT, LSTRIDE, 128, bufB);
  gemm16<128, 7, 0, true>(p.Wp1, 128, nullptr, bufT, rawD, 28, 7, nullptr);
  if (rowOwner) {
#pragma unroll
    for (int j = 0; j < 7; ++j) rhs[j] -= rawD[lane * 28 + j];
  }

  // ================= mass net forward: raw tril -> L =================
  gemm16<7, 128, 1, false>(p.Wm1, 128, p.bm1, bufA, bufB, LSTRIDE, 128, nullptr);
  gemm16<128, 128, 1, false>(p.Wm2, 128, p.bm2, bufB, bufT, LSTRIDE, 128, nullptr);
  gemm16<128, 28, 0, false>(p.Wm3, 28, p.bm3, bufT, rawL, 28, 28, nullptr);

  float Lr[28], sig[7] = {0}, ww[7] = {0};
  if (rowOwner) {
#pragma unroll
    for (int t = 0; t < 28; ++t) Lr[t] = rawL[lane * 28 + t];
#pragma unroll
    for (int d = 0; d < 7; ++d) {
      const int pos = (d * (d + 1)) / 2 + d;
      const float z = Lr[pos];
      const float e = __expf(-fabsf(z));
      sig[d] = (z >= 0.f) ? 1.f / (1.f + e) : e / (1.f + e);    // softplus'
      Lr[pos] = fmaxf(z, 0.f) + log1pf(e) + 1e-3f;              // softplus(z)+1e-3
    }
    // w = L^T v
#pragma unroll
    for (int kk = 0; kk < 7; ++kk) {
      float s = 0.f;
#pragma unroll
      for (int i = kk; i < 7; ++i) s += Lr[(i * (i + 1)) / 2 + kk] * vv[i];
      ww[kk] = s;
    }
  }

  // ================= jacfwd tangents, one direction at a time =================
  // dz1/dq_k = Wm1[k,:] (batch-constant). Cv_i = sum_k v_k (dL_k w + L u_k)_i - u_i.w
  for (int kd = 0; kd < 7; ++kd) {
    for (int t = lane; t < TILE_ROWS * 128; t += 32) {          // t_h1 = W1row ⊙ (1-h1^2)
      const int m = t >> 7, c = t & 127;
      const float h = bufB[m * LSTRIDE + c];
      bufA[m * LSTRIDE + c] = p.Wm1[kd * 128 + c] * (1.f - h * h);
    }
    gemm16<128, 128, 2, false>(p.Wm2, 128, nullptr, bufA, bufA, LSTRIDE, 128, bufT);
    gemm16<128, 28, 0, false>(p.Wm3, 28, nullptr, bufA, rawD, 28, 28, nullptr);

    if (rowOwner) {
      float dl[28];
#pragma unroll
      for (int t = 0; t < 28; ++t) dl[t] = rawD[lane * 28 + t];
#pragma unroll
      for (int d = 0; d < 7; ++d) dl[(d * (d + 1)) / 2 + d] *= sig[d];
      float uu[7];
#pragma unroll
      for (int i = 0; i < 7; ++i) {                              // u = dL^T v
        float s = 0.f;
#pragma unroll
        for (int j = i; j < 7; ++j) s += dl[(j * (j + 1)) / 2 + i] * vv[j];
        uu[i] = s;
      }
      float du = 0.f;
#pragma unroll
      for (int i = 0; i < 7; ++i) {
        float pacc = 0.f;
#pragma unroll
        for (int j = 0; j <= i; ++j) {
          const int t = (i * (i + 1)) / 2 + j;
          pacc += dl[t] * ww[j] + Lr[t] * uu[j];                 // (dL w + L u)_i
        }
        Cv[i] += vv[kd] * pacc;
        du += uu[i] * ww[i];
      }
      Cv[kd] -= du;
    }
  }

  // ================= rhs = F - G - Cv ;  solve (L L^T) x = rhs =================
  if (rowOwner) {
    float b[7], y[7], x[7];
#pragma unroll
    for (int j = 0; j < 7; ++j) b[j] = rhs[j] - Cv[j];
#pragma unroll
    for (int i = 0; i < 7; ++i) {                                // L y = b
      float s = b[i];
#pragma unroll
      for (int j = 0; j < i; ++j) s -= Lr[(i * (i + 1)) / 2 + j] * y[j];
      y[i] = s / Lr[(i * (i + 1)) / 2 + i];
    }
#pragma unroll
    for (int i = 6; i >= 0; --i) {                               // L^T x = y
      float s = y[i];
#pragma unroll
      for (int j = i + 1; j < 7; ++j) s -= Lr[(j * (j + 1)) / 2 + i] * x[j];
      x[i] = s / Lr[(i * (i + 1)) / 2 + i];
    }
#pragma unroll
    for (int j = 0; j < 7; ++j) p.out[(r0 + lane) * 7 + j] = x[j];
  }
}

extern "C" void kernel_launch(void* const* d_in, const int* in_sizes, int n_in,
                              void* d_out, int out_size, void* d_ws, size_t ws_size,
                              hipStream_t stream) {
  (void)d_ws; (void)ws_size; (void)out_size;
  LrbParams p{};
  p.q = (const float*)d_in[0];
  p.v = (const float*)d_in[1];
  p.u = (const float*)d_in[2];
  // Params pytree order (dict key order) is ambiguous -> identify nets by size signature:
  //   force: W1 = 21*128 = 2688 ; mm: W3 = 128*28 = 3584 ; pot: W3 = 128*1 = 128
  int idx = 3;
  for (int g = 0; g < 3 && idx + 5 < n_in; ++g, idx += 6) {
    const float* w1 = (const float*)d_in[idx];
    const float* b1 = (const float*)d_in[idx + 1];
    const float* w2 = (const float*)d_in[idx + 2];
    const float* b2 = (const float*)d_in[idx + 3];
    const float* w3 = (const float*)d_in[idx + 4];
    const float* b3 = (const float*)d_in[idx + 5];
    if (in_sizes[idx] == 21 * 128) {
      p.Wf1 = w1; p.bf1 = b1; p.Wf2 = w2; p.bf2 = b2; p.Wf3 = w3; p.bf3 = b3;
    } else if (in_sizes[idx + 4] == 128 * 28) {
      p.Wm1 = w1; p.bm1 = b1; p.Wm2 = w2; p.bm2 = b2; p.Wm3 = w3; p.bm3 = b3;
    } else {
      p.Wp1 = w1; p.bp1 = b1; p.Wp2 = w2; p.bp2 = b2; p.Wp3 = w3; p.bp3 = b3;
    }
  }
  p.out = (float*)d_out;
  p.B   = in_sizes[0] / 7;
  const int nblk = (p.B + TILE_ROWS - 1) / TILE_ROWS;
  lrb_fused_kernel<<<nblk, 32, 0, stream>>>(p);
}